// BatchTopKFilter_7567732376178
// MI455X (gfx1250) — compile-verified
//
#include <hip/hip_runtime.h>
#include <stdint.h>

// ---------------------------------------------------------------------------
// BatchTopK (global top k*B over 4096x24576 fp32).
// Threshold found by radix refinement of the monotone uint32 key; exact
// reference semantics (lowest-index-first tie break) via radix-select of the
// E-th smallest index among exact-threshold ties. Final streaming mask pass.
//
// CDNA5 paths: global_load_async_to_lds_b128 (+ s_wait_asynccnt) staging in
// the histogram pass, replicated LDS histograms (ds_add), nontemporal (NT)
// cache policy on all 400MB streams (>> 192MB L2, zero reuse).
//
// Two host-selected plans (deterministic, chosen from ws_size only):
//   fast (ws >= ~17MB): hist1 -> scan -> compact matching elements (~490K)
//                       -> single-block resolve from compact set -> mask
//                       traffic ~1.61GB  (~69us @ 23.3TB/s)
//   slow (small ws):    hist1 -> scan -> hist2 -> scan -> tiny compact
//                       -> resolve -> mask       traffic ~2.0GB (~86us)
// ---------------------------------------------------------------------------

#define WG 256u

typedef __attribute__((ext_vector_type(4))) float vf4;

// ws layout (32-bit words)
#define OFF_H1    0       // 2048 words: pass-1 histogram
#define OFF_H2    2048    // 2048 words: pass-2 histogram (slow path)
#define OFF_CTRL  4096    // 64 words: control block
#define OFF_CAND  4160    // candidate buffer: pairs
// control indices
#define C_B1      0
#define C_R2      1
#define C_B2      2
#define C_R3      3
#define C_KV      4
#define C_E       5
#define C_IDXT    6
#define C_TICKET  7

typedef __attribute__((address_space(3))) void lds_void_t;

__device__ __forceinline__ unsigned lds_off(const void* p) {
  return (unsigned)(uintptr_t)(lds_void_t*)(void*)p;
}

__device__ __forceinline__ void async_copy_b128(unsigned dst_lds, const float* src) {
  asm volatile("global_load_async_to_lds_b128 %0, %1, off th:TH_LOAD_NT"
               :: "v"(dst_lds), "v"(src) : "memory");
}

__device__ __forceinline__ void wait_async0() {
#if __has_builtin(__builtin_amdgcn_s_wait_asynccnt)
  __builtin_amdgcn_s_wait_asynccnt(0);
#else
  asm volatile("s_wait_asynccnt 0x0" ::: "memory");
#endif
  asm volatile("" ::: "memory");
}
__device__ __forceinline__ void wait_async1() {
#if __has_builtin(__builtin_amdgcn_s_wait_asynccnt)
  __builtin_amdgcn_s_wait_asynccnt(1);
#else
  asm volatile("s_wait_asynccnt 0x1" ::: "memory");
#endif
  asm volatile("" ::: "memory");
}

// monotone key: larger float -> larger unsigned key
__device__ __forceinline__ unsigned keyOf(float x) {
  unsigned u = __float_as_uint(x);
  unsigned m = (unsigned)((int)u >> 31) | 0x80000000u;
  return u ^ m;
}

// ---------------------------------------------------------------------------
__global__ void k_zero(unsigned* p, unsigned nw) {
  unsigned i = blockIdx.x * blockDim.x + threadIdx.x;
  if (i < nw) p[i] = 0u;
}

// ---------------------------------------------------------------------------
// Pass 1: 2048-bin histogram of key>>21, async-to-LDS double-buffered input,
// 8-way replicated LDS histogram to cut same-bin ds_add serialization.
__global__ __launch_bounds__(256) void k_hist1(const float* __restrict__ in,
                                               size_t n4,
                                               unsigned* __restrict__ hist1) {
  __shared__ __align__(16) float stage[2][WG * 4];   // 8 KB
  __shared__ unsigned lh[8][2048];                   // 64 KB
  const unsigned tid = threadIdx.x;
  for (unsigned i = tid; i < 8u * 2048u; i += WG) ((unsigned*)lh)[i] = 0u;
  __syncthreads();

  const unsigned rep = tid & 7u;
  const size_t numTiles = n4 / WG;
  const size_t stride = (size_t)gridDim.x;
  size_t t = (size_t)blockIdx.x;
  unsigned cur = 0u;

  if (t < numTiles)
    async_copy_b128(lds_off(&stage[0][tid * 4]), in + (t * WG + tid) * 4);

  for (; t < numTiles; t += stride) {
    size_t tn = t + stride;
    if (tn < numTiles) {
      async_copy_b128(lds_off(&stage[cur ^ 1u][tid * 4]), in + (tn * WG + tid) * 4);
      wait_async1();   // oldest (current tile) done
    } else {
      wait_async0();
    }
    vf4 v = *reinterpret_cast<const vf4*>(&stage[cur][tid * 4]);
    atomicAdd(&lh[rep][keyOf(v.x) >> 21], 1u);
    atomicAdd(&lh[rep][keyOf(v.y) >> 21], 1u);
    atomicAdd(&lh[rep][keyOf(v.z) >> 21], 1u);
    atomicAdd(&lh[rep][keyOf(v.w) >> 21], 1u);
    cur ^= 1u;
  }

  // tail (n4 % WG) — no-op for this problem size
  if (blockIdx.x == 0) {
    for (size_t i = numTiles * WG + tid; i < n4; i += WG) {
      vf4 v = __builtin_nontemporal_load(reinterpret_cast<const vf4*>(in) + i);
      atomicAdd(&lh[rep][keyOf(v.x) >> 21], 1u);
      atomicAdd(&lh[rep][keyOf(v.y) >> 21], 1u);
      atomicAdd(&lh[rep][keyOf(v.z) >> 21], 1u);
      atomicAdd(&lh[rep][keyOf(v.w) >> 21], 1u);
    }
  }
  __syncthreads();
  for (unsigned b = tid; b < 2048u; b += WG) {
    unsigned s = 0u;
#pragma unroll
    for (int r = 0; r < 8; ++r) s += lh[r][b];
    if (s) atomicAdd(&hist1[b], s);
  }
}

// ---------------------------------------------------------------------------
// Suffix-scan a 2048-bin histogram, find threshold bin + residual rank.
// mode 0: R = (*kptr)*Bdim, writes C_B1/C_R2.  mode 1: R = ctrl[C_R2], writes C_B2/C_R3.
__global__ __launch_bounds__(1024) void k_scan2048(const unsigned* __restrict__ hist,
                                                   unsigned* __restrict__ ctrl,
                                                   const int* __restrict__ kptr,
                                                   unsigned Bdim, int mode) {
  __shared__ unsigned s0[2048], s1[2048];
  const unsigned tid = threadIdx.x;
  s0[tid]        = hist[tid];
  s0[tid + 1024] = hist[tid + 1024];
  __syncthreads();
  unsigned* src = s0; unsigned* dst = s1;
  for (int off = 1; off < 2048; off <<= 1) {
    for (unsigned i = tid; i < 2048u; i += 1024u) {
      unsigned v = src[i];
      if (i + off < 2048u) v += src[i + off];
      dst[i] = v;
    }
    __syncthreads();
    unsigned* tmp = src; src = dst; dst = tmp;
  }
  unsigned R = (mode == 0) ? (unsigned)(*kptr) * Bdim : ctrl[C_R2];
  for (unsigned i = tid; i < 2048u; i += 1024u) {
    unsigned Si = src[i];
    unsigned Sn = (i + 1u < 2048u) ? src[i + 1] : 0u;
    if (Si >= R && Sn < R) {
      if (mode == 0) { ctrl[C_B1] = i; ctrl[C_R2] = R - Sn; }
      else           { ctrl[C_B2] = i; ctrl[C_R3] = R - Sn; }
    }
  }
}

// ---------------------------------------------------------------------------
// FAST PATH pass 2: compact (idx, low21) of elements whose top-11 bits == b1.
__global__ __launch_bounds__(256) void k_collect1(const float* __restrict__ in, size_t n4,
                                                  unsigned* __restrict__ ctrl,
                                                  unsigned* __restrict__ cand, unsigned cap) {
  const unsigned b1 = ctrl[C_B1];
  size_t i = (size_t)blockIdx.x * blockDim.x + threadIdx.x;
  size_t stride = (size_t)gridDim.x * blockDim.x;
  for (; i < n4; i += stride) {
    vf4 v = __builtin_nontemporal_load(reinterpret_cast<const vf4*>(in) + i);
    const float xs[4] = {v.x, v.y, v.z, v.w};
#pragma unroll
    for (int j = 0; j < 4; ++j) {
      unsigned kk = keyOf(xs[j]);
      if ((kk >> 21) == b1) {
        unsigned tk = atomicAdd(&ctrl[C_TICKET], 1u);
        if (tk < cap) {
          cand[2u * tk]     = (unsigned)(i * 4 + j);
          cand[2u * tk + 1] = kk & 0x1FFFFFu;
        }
      }
    }
  }
}

// ---------------------------------------------------------------------------
__device__ __forceinline__ unsigned blockReduceSum1024(unsigned v, unsigned* red) {
#pragma unroll
  for (int o = 16; o > 0; o >>= 1) v += __shfl_down(v, o);
  unsigned lane = threadIdx.x & 31u, wid = threadIdx.x >> 5;
  if (lane == 0) red[wid] = v;
  __syncthreads();
  if (wid == 0) {
    unsigned r = red[lane];
#pragma unroll
    for (int o = 16; o > 0; o >>= 1) r += __shfl_down(r, o);
    if (lane == 0) red[0] = r;
  }
  __syncthreads();
  unsigned total = red[0];
  __syncthreads();
  return total;
}

// FAST PATH resolve: from compacted (idx, low21) set resolve bits [20:10],
// [9:0], quota E and the E-th smallest index among exact ties.
__global__ __launch_bounds__(1024) void k_resolve_big(unsigned* __restrict__ ctrl,
                                                      const unsigned* __restrict__ cand,
                                                      unsigned cap) {
  __shared__ unsigned hA[2048], hB[2048];
  __shared__ unsigned eq[2048];
  __shared__ unsigned red[32];
  __shared__ unsigned sh_b2, sh_R3, sh_b3, sh_E, sh_prefix, sh_ne;
  const unsigned tid = threadIdx.x;
  const unsigned M  = min(ctrl[C_TICKET], cap);
  const unsigned R2 = ctrl[C_R2];

  if (tid == 0) { sh_b2 = 0u; sh_R3 = 1u; sh_b3 = 0u; sh_E = 1u; sh_prefix = 0u; sh_ne = 0u; }
  hA[tid] = 0u; hA[tid + 1024] = 0u;
  __syncthreads();

  // phase A: histogram bits [20:10]
  for (unsigned i = tid; i < M; i += 1024u) atomicAdd(&hA[cand[2u * i + 1] >> 10], 1u);
  __syncthreads();
  unsigned* src = hA; unsigned* dst = hB;
  for (int off = 1; off < 2048; off <<= 1) {
    for (unsigned i = tid; i < 2048u; i += 1024u) {
      unsigned v = src[i];
      if (i + off < 2048u) v += src[i + off];
      dst[i] = v;
    }
    __syncthreads();
    unsigned* t = src; src = dst; dst = t;
  }
  for (unsigned i = tid; i < 2048u; i += 1024u) {
    unsigned Si = src[i], Sn = (i + 1u < 2048u) ? src[i + 1] : 0u;
    if (Si >= R2 && Sn < R2) { sh_b2 = i; sh_R3 = R2 - Sn; ctrl[C_B2] = i; ctrl[C_R3] = R2 - Sn; }
  }
  __syncthreads();
  const unsigned b2 = sh_b2, R3 = sh_R3;

  // phase B: histogram bits [9:0] of entries matching b2
  hA[tid] = 0u;
  __syncthreads();
  for (unsigned i = tid; i < M; i += 1024u) {
    unsigned lo = cand[2u * i + 1];
    if ((lo >> 10) == b2) atomicAdd(&hA[lo & 0x3FFu], 1u);
  }
  __syncthreads();
  src = hA; dst = hB;
  for (int off = 1; off < 1024; off <<= 1) {
    unsigned v = src[tid];
    if (tid + off < 1024u) v += src[tid + off];
    dst[tid] = v;
    __syncthreads();
    unsigned* t = src; src = dst; dst = t;
  }
  {
    unsigned Si = src[tid], Sn = (tid < 1023u) ? src[tid + 1] : 0u;
    if (Si >= R3 && Sn < R3) {
      sh_b3 = tid; sh_E = R3 - Sn;
      ctrl[C_KV] = (ctrl[C_B1] << 21) | (b2 << 10) | tid;
      ctrl[C_E]  = R3 - Sn;
    }
  }
  __syncthreads();
  const unsigned b3 = sh_b3;
  const unsigned lo21 = (b2 << 10) | b3;

  // phase C: compact exact-tie indices into LDS
  for (unsigned i = tid; i < M; i += 1024u) {
    if (cand[2u * i + 1] == lo21) {
      unsigned t = atomicAdd(&sh_ne, 1u);
      if (t < 2048u) eq[t] = cand[2u * i];
    }
  }
  __syncthreads();
  const unsigned ne = min(sh_ne, 2048u);

  // phase D: radix-select E-th smallest index among ties (all in LDS)
  for (int bit = 30; bit >= 0; --bit) {
    unsigned pfx = sh_prefix;
    unsigned c = 0u;
    for (unsigned i = tid; i < ne; i += 1024u) {
      unsigned idx = eq[i];
      if ((idx >> (bit + 1)) == pfx && ((idx >> bit) & 1u) == 0u) c++;
    }
    unsigned cnt0 = blockReduceSum1024(c, red);
    if (tid == 0) {
      if (sh_E <= cnt0) sh_prefix = pfx << 1;
      else { sh_E -= cnt0; sh_prefix = (pfx << 1) | 1u; }
    }
    __syncthreads();
  }
  if (tid == 0) ctrl[C_IDXT] = sh_prefix;
}

// ---------------------------------------------------------------------------
// SLOW PATH pass 2: histogram bits [20:10] of elements whose top-11 bits == b1.
__global__ __launch_bounds__(256) void k_hist2(const float* __restrict__ in, size_t n4,
                                               const unsigned* __restrict__ ctrl,
                                               unsigned* __restrict__ hist2) {
  const unsigned b1 = ctrl[C_B1];
  size_t i = (size_t)blockIdx.x * blockDim.x + threadIdx.x;
  size_t stride = (size_t)gridDim.x * blockDim.x;
  for (; i < n4; i += stride) {
    vf4 v = __builtin_nontemporal_load(reinterpret_cast<const vf4*>(in) + i);
    unsigned k0 = keyOf(v.x), k1 = keyOf(v.y), k2 = keyOf(v.z), k3 = keyOf(v.w);
    if ((k0 >> 21) == b1) atomicAdd(&hist2[(k0 >> 10) & 0x7FFu], 1u);
    if ((k1 >> 21) == b1) atomicAdd(&hist2[(k1 >> 10) & 0x7FFu], 1u);
    if ((k2 >> 21) == b1) atomicAdd(&hist2[(k2 >> 10) & 0x7FFu], 1u);
    if ((k3 >> 21) == b1) atomicAdd(&hist2[(k3 >> 10) & 0x7FFu], 1u);
  }
}

// SLOW PATH pass 3: compact (idx, low10) of elements matching 22-bit prefix.
__global__ __launch_bounds__(256) void k_collect(const float* __restrict__ in, size_t n4,
                                                 unsigned* __restrict__ ctrl,
                                                 unsigned* __restrict__ cand, unsigned cap) {
  const unsigned p21 = (ctrl[C_B1] << 11) | ctrl[C_B2];
  size_t i = (size_t)blockIdx.x * blockDim.x + threadIdx.x;
  size_t stride = (size_t)gridDim.x * blockDim.x;
  for (; i < n4; i += stride) {
    vf4 v = __builtin_nontemporal_load(reinterpret_cast<const vf4*>(in) + i);
    const float xs[4] = {v.x, v.y, v.z, v.w};
#pragma unroll
    for (int j = 0; j < 4; ++j) {
      unsigned kk = keyOf(xs[j]);
      if ((kk >> 10) == p21) {
        unsigned tk = atomicAdd(&ctrl[C_TICKET], 1u);
        if (tk < cap) {
          cand[2u * tk]     = (unsigned)(i * 4 + j);
          cand[2u * tk + 1] = kk & 0x3FFu;
        }
      }
    }
  }
}

// SLOW PATH resolve: exact Kv, quota E, E-th smallest tie index.
__global__ __launch_bounds__(1024) void k_resolve(unsigned* __restrict__ ctrl,
                                                  const unsigned* __restrict__ cand,
                                                  unsigned cap) {
  __shared__ unsigned h[1024], h2[1024];
  __shared__ unsigned red[32];
  __shared__ unsigned sh_b3, sh_E, sh_prefix;
  const unsigned tid = threadIdx.x;
  const unsigned M = min(ctrl[C_TICKET], cap);

  if (tid == 0) { sh_b3 = 0u; sh_E = 1u; sh_prefix = 0u; }
  h[tid] = 0u;
  __syncthreads();
  for (unsigned i = tid; i < M; i += 1024u) atomicAdd(&h[cand[2u * i + 1]], 1u);
  __syncthreads();

  unsigned* src = h; unsigned* dst = h2;
  for (int off = 1; off < 1024; off <<= 1) {
    unsigned v = src[tid];
    if (tid + off < 1024u) v += src[tid + off];
    dst[tid] = v;
    __syncthreads();
    unsigned* tmp = src; src = dst; dst = tmp;
  }
  const unsigned R3 = ctrl[C_R3];
  {
    unsigned Si = src[tid];
    unsigned Sn = (tid < 1023u) ? src[tid + 1] : 0u;
    if (Si >= R3 && Sn < R3) {
      sh_b3 = tid;
      sh_E = R3 - Sn;
      ctrl[C_KV] = (ctrl[C_B1] << 21) | (ctrl[C_B2] << 10) | tid;
      ctrl[C_E]  = R3 - Sn;
    }
  }
  __syncthreads();
  const unsigned b3 = sh_b3;

  for (int bit = 30; bit >= 0; --bit) {
    unsigned pfx = sh_prefix;
    unsigned c = 0u;
    for (unsigned i = tid; i < M; i += 1024u) {
      unsigned idx = cand[2u * i], lo = cand[2u * i + 1];
      if (lo == b3 && (idx >> (bit + 1)) == pfx && ((idx >> bit) & 1u) == 0u) c++;
    }
    unsigned cnt0 = blockReduceSum1024(c, red);
    if (tid == 0) {
      if (sh_E <= cnt0) sh_prefix = pfx << 1;
      else { sh_E -= cnt0; sh_prefix = (pfx << 1) | 1u; }
    }
    __syncthreads();
  }
  if (tid == 0) ctrl[C_IDXT] = sh_prefix;
}

// ---------------------------------------------------------------------------
// Final streaming mask: keep iff key > Kv, or key == Kv && idx <= idxThresh.
__global__ __launch_bounds__(256) void k_mask(const float* __restrict__ in,
                                              float* __restrict__ out, size_t n4,
                                              const unsigned* __restrict__ ctrl) {
  const unsigned Kv = ctrl[C_KV];
  const unsigned idxT = ctrl[C_IDXT];
  size_t i = (size_t)blockIdx.x * blockDim.x + threadIdx.x;
  size_t stride = (size_t)gridDim.x * blockDim.x;
  for (; i < n4; i += stride) {
    vf4 v = __builtin_nontemporal_load(reinterpret_cast<const vf4*>(in) + i);
    size_t base = i * 4;
    vf4 o;
    unsigned k0 = keyOf(v.x), k1 = keyOf(v.y), k2 = keyOf(v.z), k3 = keyOf(v.w);
    o.x = (k0 > Kv || (k0 == Kv && (unsigned)(base + 0) <= idxT)) ? v.x : 0.0f;
    o.y = (k1 > Kv || (k1 == Kv && (unsigned)(base + 1) <= idxT)) ? v.y : 0.0f;
    o.z = (k2 > Kv || (k2 == Kv && (unsigned)(base + 2) <= idxT)) ? v.z : 0.0f;
    o.w = (k3 > Kv || (k3 == Kv && (unsigned)(base + 3) <= idxT)) ? v.w : 0.0f;
    __builtin_nontemporal_store(o, reinterpret_cast<vf4*>(out) + i);
  }
}

// ---------------------------------------------------------------------------
extern "C" void kernel_launch(void* const* d_in, const int* in_sizes, int n_in,
                              void* d_out, int out_size, void* d_ws, size_t ws_size,
                              hipStream_t stream) {
  (void)n_in; (void)out_size;
  const float* in   = (const float*)d_in[0];
  const int*   kptr = (const int*)d_in[1];
  float*       out  = (float*)d_out;

  const size_t n  = (size_t)in_sizes[0];
  const size_t n4 = n / 4;                 // divisible for this problem
  const unsigned Bdim = 4096u;             // batch dim per reference (R = k*B)

  unsigned* ws    = (unsigned*)d_ws;
  unsigned* hist1 = ws + OFF_H1;
  unsigned* hist2 = ws + OFF_H2;
  unsigned* ctrl  = ws + OFF_CTRL;
  unsigned* cand  = ws + OFF_CAND;

  const size_t wsWords = ws_size / 4;

  // zero hist1 + hist2 + ctrl (ws is not re-poisoned between replays)
  k_zero<<<(OFF_CAND + 255) / 256, 256, 0, stream>>>(ws, (unsigned)OFF_CAND);
  k_hist1<<<2048, 256, 0, stream>>>(in, n4, hist1);
  k_scan2048<<<1, 1024, 0, stream>>>(hist1, ctrl, kptr, Bdim, 0);

  // Fast path if scratch can hold >= 2M candidate pairs (4x margin over the
  // ~490K expected matches of the pass-1 threshold bin).
  const size_t FAST_MIN_ENTRIES = (size_t)2u << 20;
  if (wsWords >= (size_t)OFF_CAND + 2 * FAST_MIN_ENTRIES) {
    size_t c = (wsWords - OFF_CAND) / 2;
    if (c > ((size_t)8u << 20)) c = (size_t)8u << 20;
    unsigned cap = (unsigned)c;
    k_collect1<<<2048, 256, 0, stream>>>(in, n4, ctrl, cand, cap);
    k_resolve_big<<<1, 1024, 0, stream>>>(ctrl, cand, cap);
  } else {
    unsigned cap = 1024u;
    if (wsWords > (size_t)(OFF_CAND + 2048)) {
      size_t c = (wsWords - OFF_CAND) / 2;
      if (c > 32768) c = 32768;
      cap = (unsigned)c;
    }
    k_hist2<<<2048, 256, 0, stream>>>(in, n4, ctrl, hist2);
    k_scan2048<<<1, 1024, 0, stream>>>(hist2, ctrl, kptr, Bdim, 1);
    k_collect<<<2048, 256, 0, stream>>>(in, n4, ctrl, cand, cap);
    k_resolve<<<1, 1024, 0, stream>>>(ctrl, cand, cap);
  }
  k_mask<<<8192, 256, 0, stream>>>(in, out, n4, ctrl);
}